// GATLayer_2697239462336
// MI455X (gfx1250) — compile-verified
//
#include <hip/hip_runtime.h>
#include <hip/hip_bf16.h>

// GAT layer for MI455X (gfx1250, wave32).
// B=512, V=128, D=256, H=4, E=2, Dh=64.
// fp32 WMMA (v_wmma_f32_16x16x4_f32) for both GEMMs; fused
// attention+softmax+aggregation per (b,h) block in LDS (96KB of 320KB WGP LDS)
// so alpha (134MB) never touches HBM. K1 reads each x-strip from HBM exactly
// once (single block spans all of N=256); K3 uses 4 accumulators per wave so
// each attention A-fragment is loaded from LDS once and used by 4 WMMAs.

constexpr int kB  = 512;
constexpr int kV  = 128;
constexpr int kD  = 256;
constexpr int kH  = 4;
constexpr int kE  = 2;
constexpr int kDh = 64;

typedef __attribute__((ext_vector_type(2))) float v2f;
typedef __attribute__((ext_vector_type(8))) float v8f;

// ---------------------------------------------------------------------------
// Kernel 1: Wx = x @ W^T.  out[m,n] = sum_k x[m,k]*W[n,k].
// Block = 256 threads = 8 waves; block tile 16(M) x 256(N).
// Wave w owns n = [w*32, w*32+32): two 16x16 tiles sharing one A fragment.
// x strip (16x256, padded stride 260) staged in LDS once per block.
// ---------------------------------------------------------------------------
__global__ void __launch_bounds__(256) gemm_wx_kernel(const float* __restrict__ x,
                                                      const float* __restrict__ W,
                                                      float* __restrict__ wx) {
    __shared__ float aLds[16 * 260];

    const int  tid = threadIdx.x;
    const long m0  = (long)blockIdx.x * 16;

    const int wave  = tid >> 5;
    const int lane  = tid & 31;
    const int lhalf = lane >> 4;     // K pair {0,1} vs {2,3}
    const int lmod  = lane & 15;     // M (A) / N (B) index
    const int n0    = wave * 32;

    const float* w0 = W + (long)(n0 + lmod) * kD;        // B[k][n]      = W[n][k]
    const float* w1 = W + (long)(n0 + 16 + lmod) * kD;   // B[k][n+16]
    // Warm L2 for this wave's W rows while the LDS stage is in flight
    // (lowers to global_prefetch_b8).
    __builtin_prefetch(w0, 0, 0);
    __builtin_prefetch(w1, 0, 0);

    // Stage the 16x256 A strip (1024 float4 loads over 256 threads).
    const float4* xv = (const float4*)(x + m0 * kD);
    for (int i = tid; i < 16 * 64; i += 256) {
        int row = i >> 6, c4 = i & 63;
        float4 val = xv[(long)row * 64 + c4];
        float* dst = &aLds[row * 260 + c4 * 4];
        dst[0] = val.x; dst[1] = val.y; dst[2] = val.z; dst[3] = val.w;
    }
    __syncthreads();

    v8f c0 = {}, c1 = {};
    for (int k0 = 0; k0 < kD; k0 += 4) {
        const int k = k0 + 2 * lhalf;
        v2f af, b0, b1;
        af.x = aLds[lmod * 260 + k];
        af.y = aLds[lmod * 260 + k + 1];
        b0.x = w0[k]; b0.y = w0[k + 1];
        b1.x = w1[k]; b1.y = w1[k + 1];
        c0 = __builtin_amdgcn_wmma_f32_16x16x4_f32(false, af, false, b0,
                                                   (short)0, c0, false, false);
        c1 = __builtin_amdgcn_wmma_f32_16x16x4_f32(false, af, false, b1,
                                                   (short)0, c1, false, false);
    }

#pragma unroll
    for (int v = 0; v < 8; ++v) {
        const int m = v + 8 * lhalf;       // C layout: VGPR v -> M = v (+8 hi half)
        wx[(m0 + m) * kD + n0 + lmod]      = c0[v];
        wx[(m0 + m) * kD + n0 + 16 + lmod] = c1[v];
    }
}

// ---------------------------------------------------------------------------
// Kernel 2: s_src[b,v,h] = Wx[b,v,h,:] . a[0:64]; s_dst with a[64:128].
// One wave32 per (b,v,h) row of 64 elements; shfl_xor tree reduction.
// ---------------------------------------------------------------------------
__global__ void __launch_bounds__(256) scores_kernel(const float* __restrict__ wx,
                                                     const float* __restrict__ a,
                                                     float* __restrict__ s_src,
                                                     float* __restrict__ s_dst) {
    const int  wave = threadIdx.x >> 5;
    const int  lane = threadIdx.x & 31;
    const long r    = (long)blockIdx.x * 8 + wave;     // < B*V*H
    const long bv   = r >> 2;                          // H == 4
    const int  h    = (int)(r & 3);

    const float* base = wx + bv * kD + h * kDh;
    const float2 wv = *(const float2*)(base + lane * 2);
    const float2 as = *(const float2*)(a + lane * 2);
    const float2 ad = *(const float2*)(a + kDh + lane * 2);

    float ps = wv.x * as.x + wv.y * as.y;
    float pd = wv.x * ad.x + wv.y * ad.y;
#pragma unroll
    for (int off = 16; off > 0; off >>= 1) {
        ps += __shfl_xor(ps, off, 32);
        pd += __shfl_xor(pd, off, 32);
    }
    if (lane == 0) {
        s_src[r] = ps;
        s_dst[r] = pd;
    }
}

// ---------------------------------------------------------------------------
// Kernel 3: fused attention per (b,h) block.
//   eLds[i][j]  = leakyrelu(s_src[b,i,h]+s_dst[b,j,h]+ef[i,j,h]) or -inf
//   softmax over i (per column j), then
//   out[b,j,h,:]= ELU( sum_i alpha[i][j] * Wx[b,i,h,:] ), masked.
// Aggregation = alpha^T (128x128) x Wx_h (128x64) with WMMA f32 16x16x4;
// each wave holds 4 accumulators (j-tile x 4 d-tiles), so one A fragment
// feeds 4 WMMAs. Dynamic LDS: 128*128 + 128*64 floats = 96KB.
// ---------------------------------------------------------------------------
__global__ void __launch_bounds__(256) attn_agg_kernel(const float* __restrict__ wx,
                                                       const int* __restrict__ adj,
                                                       const float* __restrict__ edge_attr,
                                                       const unsigned char* __restrict__ mask,
                                                       const float* __restrict__ Ew,
                                                       const float* __restrict__ s_src,
                                                       const float* __restrict__ s_dst,
                                                       float* __restrict__ out) {
    extern __shared__ float smem[];
    float* eLds  = smem;               // [128][128]
    float* wxLds = smem + kV * kV;     // [128][64]

    const int tid = threadIdx.x;
    const int b   = blockIdx.x >> 2;   // H == 4
    const int h   = blockIdx.x & 3;
    const float NEG_INF = -__builtin_inff();

    // Stage Wx head slice: Wx[b, i, h*64 + d], i in [0,128), d in [0,64).
    const float* wbase = wx + ((long)b * kV) * kD + h * kDh;
    for (int i4 = tid; i4 < kV * 16; i4 += 256) {
        int row = i4 >> 4, c4 = i4 & 15;
        float4 val = *(const float4*)(wbase + (long)row * kD + c4 * 4);
        float* dst = &wxLds[row * kDh + c4 * 4];
        dst[0] = val.x; dst[1] = val.y; dst[2] = val.z; dst[3] = val.w;
    }

    // Attention logits with leaky-ReLU and validity mask.
    const float ew0 = Ew[h * kE + 0];
    const float ew1 = Ew[h * kE + 1];
    for (int idx = tid; idx < kV * kV; idx += 256) {
        const int i = idx >> 7, j = idx & 127;
        const bool valid = (adj[i * kV + j] != 0) &&
                           (mask[b * kV + i] == 0) && (mask[b * kV + j] == 0);
        const float ef = edge_attr[(i * kV + j) * kE + 0] * ew0 +
                         edge_attr[(i * kV + j) * kE + 1] * ew1;
        float ev = s_src[((long)b * kV + i) * kH + h] +
                   s_dst[((long)b * kV + j) * kH + h] + ef;
        ev = (ev >= 0.0f) ? ev : 0.2f * ev;
        eLds[idx] = valid ? ev : NEG_INF;
    }
    __syncthreads();

    // Column softmax over i (threads 0..127, one column each).
    if (tid < kV) {
        const int j = tid;
        float m = NEG_INF;
        for (int i = 0; i < kV; ++i) m = fmaxf(m, eLds[i * kV + j]);
        if (m == NEG_INF) m = 0.0f;                 // "not finite -> 0"
        float s = 0.0f;
        for (int i = 0; i < kV; ++i) {
            const float p = __expf(eLds[i * kV + j] - m);  // exp(-inf)=0 covers invalid
            eLds[i * kV + j] = p;
            s += p;
        }
        const float inv = (s == 0.0f) ? 1.0f : 1.0f / s;   // denom==0 -> divide by 1
        for (int i = 0; i < kV; ++i) eLds[i * kV + j] *= inv;
    }
    __syncthreads();

    // Aggregation: out[j][d] = sum_i alpha[i][j] * wxLds[i][d].
    // A[m=j][k=i] = eLds[i*128+j], B[k=i][n=d] = wxLds[i*64+d].
    const int wave  = tid >> 5;
    const int lane  = tid & 31;
    const int lhalf = lane >> 4;
    const int lmod  = lane & 15;
    const int j0    = wave * 16;                    // 8 waves x 16 rows = 128 j
    const unsigned char* mrow = mask + (long)b * kV;

    v8f acc[4] = {};
    for (int k0 = 0; k0 < kV; k0 += 4) {
        const int k = k0 + 2 * lhalf;
        v2f af;
        af.x = eLds[(k)     * kV + (j0 + lmod)];
        af.y = eLds[(k + 1) * kV + (j0 + lmod)];
#pragma unroll
        for (int dt = 0; dt < 4; ++dt) {            // 4 d-tiles share this A frag
            v2f bf;
            bf.x = wxLds[(k)     * kDh + (dt * 16 + lmod)];
            bf.y = wxLds[(k + 1) * kDh + (dt * 16 + lmod)];
            acc[dt] = __builtin_amdgcn_wmma_f32_16x16x4_f32(false, af, false, bf,
                                                            (short)0, acc[dt],
                                                            false, false);
        }
    }

#pragma unroll
    for (int dt = 0; dt < 4; ++dt) {
        const int n0 = dt * 16;
#pragma unroll
        for (int v = 0; v < 8; ++v) {
            const int j = j0 + v + 8 * lhalf;
            const int d = n0 + lmod;
            float hv = acc[dt][v];
            hv = (hv > 0.0f) ? hv : (__expf(hv) - 1.0f);   // ELU
            out[((long)b * kV + j) * kD + h * kDh + d] = mrow[j] ? 0.0f : hv;
        }
    }
}

// ---------------------------------------------------------------------------
// Launch wiring. Workspace layout (floats):
//   wx     : B*V*D         = 16,777,216
//   s_src  : B*V*H         =    262,144
//   s_dst  : B*V*H         =    262,144
// Total ~69.2 MB. All scratch fully rewritten before use each call.
// Inputs (setup_inputs order): x, adj, edge_attr, mask, W, a, Ew.
// ---------------------------------------------------------------------------
extern "C" void kernel_launch(void* const* d_in, const int* in_sizes, int n_in,
                              void* d_out, int out_size, void* d_ws, size_t ws_size,
                              hipStream_t stream) {
    const float*         x         = (const float*)d_in[0];
    const int*           adj       = (const int*)d_in[1];
    const float*         edge_attr = (const float*)d_in[2];
    const unsigned char* mask      = (const unsigned char*)d_in[3];
    const float*         W         = (const float*)d_in[4];
    const float*         a         = (const float*)d_in[5];
    const float*         Ew        = (const float*)d_in[6];
    float*               out       = (float*)d_out;

    float* wx    = (float*)d_ws;
    float* s_src = wx + (size_t)kB * kV * kD;
    float* s_dst = s_src + (size_t)kB * kV * kH;

    // K1: projection GEMM; one block per 16-row M strip, full N=256 per block.
    gemm_wx_kernel<<<(kB * kV) / 16, 256, 0, stream>>>(x, W, wx);

    // K2: attention scores (8 wave-rows per block).
    scores_kernel<<<(kB * kV * kH) / 8, 256, 0, stream>>>(wx, a, s_src, s_dst);

    // K3: fused attention + aggregation, one block per (b,h), 96KB dynamic LDS.
    const size_t smemBytes = (size_t)(kV * kV + kV * kDh) * sizeof(float);
    attn_agg_kernel<<<kB * kH, 256, smemBytes, stream>>>(wx, adj, edge_attr, mask,
                                                         Ew, s_src, s_dst, out);
}